// NNCELoss_71330816852649
// MI455X (gfx1250) — compile-verified
//
#include <hip/hip_runtime.h>
#include <hip/hip_bf16.h>
#include <stdint.h>

#define EPSF   1e-7f
#define LOG2E  1.44269504088896340736f
#define LN2    0.69314718055994530942f

typedef __attribute__((ext_vector_type(2))) float v2f;
typedef __attribute__((ext_vector_type(8))) float v8f;

// D = A(16x4 f32) * ones(4x16) + C : every column of D holds the row-sums of A.
// A layout (ISA 7.12.2): lanes 0-15 hold K=0,1 in VGPR0,1; lanes 16-31 hold K=2,3.
__device__ __forceinline__ v8f wmma_rowsum_acc(v2f a, v8f acc) {
  v2f ones = {1.0f, 1.0f};
  return __builtin_amdgcn_wmma_f32_16x16x4_f32(
      /*neg_a=*/false, a, /*neg_b=*/false, ones,
      /*c_mod=*/(short)0, acc, /*reuse_a=*/false, /*reuse_b=*/false);
}

// ---------------------------------------------------------------------------
// Pass 1: per-row sum of squares -> norm (+eps). Coalesced b128 sweep.
// ---------------------------------------------------------------------------
__global__ __launch_bounds__(256) void nnce_pass1(
    const float* __restrict__ pred, float* __restrict__ norms_ws,
    float* __restrict__ norms_out, int C)
{
  __shared__ float red[8];
  const int row = blockIdx.x;
  const float* p = pred + (size_t)row * (size_t)C;
  const float4* p4 = (const float4*)p;
  const int n4 = C >> 2;
  float ss = 0.f;
  for (int i = threadIdx.x; i < n4; i += 256) {
    float4 v = p4[i];
    ss += v.x * v.x + v.y * v.y + v.z * v.z + v.w * v.w;
  }
  for (int i = (n4 << 2) + threadIdx.x; i < C; i += 256) ss += p[i] * p[i];
  #pragma unroll
  for (int off = 16; off > 0; off >>= 1) ss += __shfl_xor(ss, off, 32);
  if ((threadIdx.x & 31) == 0) red[threadIdx.x >> 5] = ss;
  __syncthreads();
  if (threadIdx.x == 0) {
    float s = 0.f;
    #pragma unroll
    for (int w = 0; w < 8; ++w) s += red[w];
    float nm = sqrtf(s) + EPSF;
    norms_ws[row] = nm;
    norms_out[row] = nm;
  }
}

// ---------------------------------------------------------------------------
// Pass 2: per-row  S1=sum(l), Z=sum(exp(l)), Zn=sum(exp(-l)), max(l), argmax(l)
// where l = pred/(2*norm).  Internally works in log2-domain: t = l*log2(e),
// so 2^t = exp(l) maps straight onto v_exp_f32, exp(-l) = v_rcp_f32(2^t),
// and sum(t)/max(t) are rescaled by ln2 once at the end.
// 16 rows per block (WMMA M-dim), 8 waves; global->LDS staged (coalesced b128),
// LDS read in WMMA f32 16x4 A-layout, sums via v_wmma_f32_16x16x4_f32 x ones.
// ---------------------------------------------------------------------------
#define RPB    16
#define TILE_C 128
#define TPAD   132   // +4 pad -> bank = (4*row + col) % 64, conflict-free
#define WAVES  8

__global__ __launch_bounds__(256) void nnce_pass2(
    const float* __restrict__ pred, const float* __restrict__ norms,
    float* __restrict__ S1, float* __restrict__ Z, float* __restrict__ Zn,
    float* __restrict__ Mx, int* __restrict__ Am, int N, int C)
{
  __shared__ float tile[RPB * TPAD];
  __shared__ float pS1[WAVES][RPB], pZ[WAVES][RPB], pZn[WAVES][RPB], pMx[WAVES][RPB];
  __shared__ int   pAm[WAVES][RPB];

  const int t    = threadIdx.x;
  const int wave = t >> 5;
  const int lane = t & 31;
  const int r    = lane & 15;   // matrix row this lane feeds
  const int hi   = lane >> 4;   // 0 -> K0,K1 ; 1 -> K2,K3
  const int rowbase = blockIdx.x * RPB;

  const float s2 = (0.5f * LOG2E) / norms[rowbase + r];  // x*s2 = l*log2e

  v8f accT = {}; v8f accE = {}; v8f accN = {};
  float mx = -3.0e38f; int mi = 0;     // in t-domain (monotonic in l)

  // staging assignment: thread t loads 8 consecutive floats of row sr
  const int sr = t >> 4;
  const int sc = (t & 15) * 8;
  const float* gsrc = pred + (size_t)(rowbase + sr) * (size_t)C;
  const int cwb = wave * 16;           // this wave's 16-column slice of the tile

  const int cfull = (C / TILE_C) * TILE_C;

  // ---- main loop: full tiles, no per-element guards ----
  for (int c0 = 0; c0 < cfull; c0 += TILE_C) {
    {
      const float4* p = (const float4*)(gsrc + c0 + sc);
      float4 a = p[0], b = p[1];
      float4* dst = (float4*)&tile[sr * TPAD + sc];
      dst[0] = a; dst[1] = b;
      if (c0 + TILE_C < C)  // hint next tile -> global_prefetch_b8
        __builtin_prefetch(gsrc + c0 + TILE_C + sc, 0, 1);
    }
    __syncthreads();

    #pragma unroll
    for (int j = 0; j < 4; ++j) {
      const int cc = cwb + 4 * j + 2 * hi;
      const int gc = c0 + cc;
      float2 x = *(const float2*)&tile[r * TPAD + cc];
      float t0 = x.x * s2, t1 = x.y * s2;
      float e0 = __builtin_amdgcn_exp2f(t0);
      float e1 = __builtin_amdgcn_exp2f(t1);
      float n0 = __builtin_amdgcn_rcpf(e0);
      float n1 = __builtin_amdgcn_rcpf(e1);
      if (t0 > mx) { mx = t0; mi = gc; }       // strict > keeps first max
      if (t1 > mx) { mx = t1; mi = gc + 1; }
      v2f aT = {t0, t1}, aE = {e0, e1}, aN = {n0, n1};
      accT = wmma_rowsum_acc(aT, accT);
      accE = wmma_rowsum_acc(aE, accE);
      accN = wmma_rowsum_acc(aN, accN);
    }
    __syncthreads();
  }

  // ---- tail tile: per-element guards ----
  if (cfull < C) {
    const int c0 = cfull;
    for (int k = 0; k < 8; ++k) {
      int gc = c0 + sc + k;
      tile[sr * TPAD + sc + k] = (gc < C) ? gsrc[gc] : 0.0f;
    }
    __syncthreads();

    #pragma unroll
    for (int j = 0; j < 4; ++j) {
      const int cc = cwb + 4 * j + 2 * hi;
      const int gc = c0 + cc;
      float2 x = *(const float2*)&tile[r * TPAD + cc];
      float t0 = x.x * s2, t1 = x.y * s2;
      const bool ok0 = gc < C, ok1 = (gc + 1) < C;
      float e0r = __builtin_amdgcn_exp2f(t0);
      float e1r = __builtin_amdgcn_exp2f(t1);
      float n0r = __builtin_amdgcn_rcpf(e0r);
      float n1r = __builtin_amdgcn_rcpf(e1r);
      float e0 = ok0 ? e0r : 0.f, e1 = ok1 ? e1r : 0.f;
      float n0 = ok0 ? n0r : 0.f, n1 = ok1 ? n1r : 0.f;
      if (!ok0) t0 = 0.f;
      if (!ok1) t1 = 0.f;
      if (ok0 && t0 > mx) { mx = t0; mi = gc; }
      if (ok1 && t1 > mx) { mx = t1; mi = gc + 1; }
      v2f aT = {t0, t1}, aE = {e0, e1}, aN = {n0, n1};
      accT = wmma_rowsum_acc(aT, accT);
      accE = wmma_rowsum_acc(aE, accE);
      accN = wmma_rowsum_acc(aN, accN);
    }
    __syncthreads();
  }

  // lanes r and r+16 both covered row r -> combine max/argmax across the pair
  float omx = __shfl_xor(mx, 16, 32);
  int   omi = __shfl_xor(mi, 16, 32);
  if (omx > mx || (omx == mx && omi < mi)) { mx = omx; mi = omi; }
  if (lane < 16) { pMx[wave][lane] = mx; pAm[wave][lane] = mi; }

  // D layout: VGPR i -> M=i (lanes 0-15, any N) / M=i+8 (lanes 16-31)
  if (lane == 0) {
    #pragma unroll
    for (int i = 0; i < 8; ++i) {
      pS1[wave][i] = accT[i]; pZ[wave][i] = accE[i]; pZn[wave][i] = accN[i];
    }
  }
  if (lane == 16) {
    #pragma unroll
    for (int i = 0; i < 8; ++i) {
      pS1[wave][8 + i] = accT[i]; pZ[wave][8 + i] = accE[i]; pZn[wave][8 + i] = accN[i];
    }
  }
  __syncthreads();

  if (t < RPB) {
    float s1 = 0.f, z = 0.f, zn = 0.f, m = -3.0e38f; int a = 0;
    #pragma unroll
    for (int w = 0; w < WAVES; ++w) {
      s1 += pS1[w][t]; z += pZ[w][t]; zn += pZn[w][t];
      float wm = pMx[w][t]; int wa = pAm[w][t];
      if (wm > m || (wm == m && wa < a)) { m = wm; a = wa; }
    }
    const int row = rowbase + t;
    S1[row] = s1 * LN2;   // back to natural-log domain
    Z[row]  = z;
    Zn[row] = zn;
    Mx[row] = m * LN2;
    Am[row] = a;
  }
}

// ---------------------------------------------------------------------------
// Threefry2x32 (JAX's PRNG), 20 rounds.
// ---------------------------------------------------------------------------
__device__ __forceinline__ uint32_t rotl32(uint32_t v, unsigned n) {
  return (v << n) | (v >> (32u - n));
}

__device__ __forceinline__ void threefry2x32(
    uint32_t k0, uint32_t k1, uint32_t c0, uint32_t c1,
    uint32_t& o0, uint32_t& o1)
{
  const uint32_t R[8] = {13, 15, 26, 6, 17, 29, 16, 24};
  uint32_t ks[3] = {k0, k1, 0x1BD11BDAu ^ k0 ^ k1};
  uint32_t x0 = c0 + ks[0];
  uint32_t x1 = c1 + ks[1];
  #pragma unroll
  for (int i = 0; i < 20; ++i) {
    x0 += x1;
    x1 = rotl32(x1, R[i & 7]);
    x1 ^= x0;
    if ((i & 3) == 3) {
      int s = (i >> 2) + 1;
      x0 += ks[s % 3];
      x1 += ks[(s + 1) % 3] + (uint32_t)s;
    }
  }
  o0 = x0; o1 = x1;
}

// ---------------------------------------------------------------------------
// Pass 3: per-row epilogue -> V[i] = sum_c log(p+eps) - log(p_lab+eps) - log(p_neg+eps)
// ---------------------------------------------------------------------------
__global__ __launch_bounds__(256) void nnce_pass3(
    const float* __restrict__ pred, const int* __restrict__ labels,
    const float* __restrict__ norms, const float* __restrict__ S1,
    const float* __restrict__ Z, const float* __restrict__ Zn,
    const float* __restrict__ Mx, const int* __restrict__ Am,
    float* __restrict__ V, int N, int C)
{
  int i = blockIdx.x * blockDim.x + threadIdx.x;
  if (i >= N) return;

  // jax.random.key(42) = (0,42); split -> k_off, k_u (threefry over iota(4))
  uint32_t a0, a1, b0, b1;
  threefry2x32(0u, 42u, 0u, 2u, a0, a1);
  threefry2x32(0u, 42u, 1u, 3u, b0, b1);
  const uint32_t ko0 = a0, ko1 = b0, ku0 = a1, ku1 = b1;

  const uint32_t half = (uint32_t)(N >> 1);
  uint32_t bits_off, bits_u, t0, t1;
  if ((uint32_t)i < half) {
    threefry2x32(ko0, ko1, (uint32_t)i, (uint32_t)i + half, t0, t1); bits_off = t0;
    threefry2x32(ku0, ku1, (uint32_t)i, (uint32_t)i + half, t0, t1); bits_u = t0;
  } else {
    threefry2x32(ko0, ko1, (uint32_t)i - half, (uint32_t)i, t0, t1); bits_off = t1;
    threefry2x32(ku0, ku1, (uint32_t)i - half, (uint32_t)i, t0, t1); bits_u = t1;
  }
  int offset = 1 + (int)(bits_off % (uint32_t)(C - 1));
  union { uint32_t u; float f; } cvt; cvt.u = 0x3F800000u | (bits_u >> 9);
  float rand_u = cvt.f - 1.0f;

  const int lab = labels[i];
  int tgt = lab + offset; if (tgt >= C) tgt -= C;

  const float z = Z[i], s1 = S1[i], zn = Zn[i], ml = Mx[i];
  const int am = Am[i];
  const float logZ = __logf(z);
  const float max_prob = __expf(ml) / z + EPSF;   // snorm = softmax + eps
  if (rand_u <= max_prob && am != lab) tgt = am;

  const float inv2n = 0.5f / norms[i];
  const float* prow = pred + (size_t)i * (size_t)C;
  const float p_lab = __expf(prow[lab] * inv2n) / z;
  const float p_neg = __expf(prow[tgt] * inv2n) / z;
  const float loglab = __logf(p_lab + EPSF);
  const float logneg = __logf(p_neg + EPSF);

  // sum_c log(p_c+eps) = S1 - C*logZ + eps*Z*sum(exp(-l)) + O(eps^2)
  const float rowlog = s1 - (float)C * logZ + EPSF * z * zn;

  V[i] = (rowlog - loglab) - logneg;
}

// ---------------------------------------------------------------------------
// Pass 4: deterministic mean reduction -> loss = 10 * mean(V)
// ---------------------------------------------------------------------------
__global__ __launch_bounds__(256) void nnce_pass4(
    const float* __restrict__ V, float* __restrict__ out, int N)
{
  __shared__ float red[8];
  float s = 0.f;
  for (int i = threadIdx.x; i < N; i += 256) s += V[i];
  #pragma unroll
  for (int off = 16; off > 0; off >>= 1) s += __shfl_xor(s, off, 32);
  if ((threadIdx.x & 31) == 0) red[threadIdx.x >> 5] = s;
  __syncthreads();
  if (threadIdx.x == 0) {
    float tot = 0.f;
    #pragma unroll
    for (int w = 0; w < 8; ++w) tot += red[w];
    out[0] = 10.0f * (tot / (float)N);
  }
}

// ---------------------------------------------------------------------------
extern "C" void kernel_launch(void* const* d_in, const int* in_sizes, int n_in,
                              void* d_out, int out_size, void* d_ws, size_t ws_size,
                              hipStream_t stream)
{
  const float* pred  = (const float*)d_in[0];
  const int* labels  = (const int*)d_in[1];
  const int N = in_sizes[1];
  const int C = in_sizes[0] / N;

  float* out = (float*)d_out;          // out[0] = loss, out[1..N] = norms
  float* ws  = (float*)d_ws;
  float* norms = ws;
  float* S1 = ws + (size_t)N;
  float* Z  = ws + 2 * (size_t)N;
  float* Zn = ws + 3 * (size_t)N;
  float* Mx = ws + 4 * (size_t)N;
  int*   Am = (int*)(ws + 5 * (size_t)N);
  float* V  = ws + 6 * (size_t)N;

  nnce_pass1<<<N, 256, 0, stream>>>(pred, norms, out + 1, C);
  nnce_pass2<<<N / RPB, 256, 0, stream>>>(pred, norms, S1, Z, Zn, Mx, Am, N, C);
  nnce_pass3<<<(N + 255) / 256, 256, 0, stream>>>(pred, labels, norms, S1, Z, Zn,
                                                  Mx, Am, V, N, C);
  nnce_pass4<<<1, 256, 0, stream>>>(V, out, N);
}